// RoIHeads_37022618092224
// MI455X (gfx1250) — compile-verified
//
#include <hip/hip_runtime.h>
#include <math.h>

// CDNA5 (gfx1250) wave32 WMMA types: 16x16x4 F32 -> A,B are 2 VGPRs, C/D 8 VGPRs
typedef __attribute__((ext_vector_type(2))) float v2f;
typedef __attribute__((ext_vector_type(8))) float v8f;

// Exact fp32 sum of the 32 lane values of a wave using V_WMMA_F32_16X16X4_F32.
// Stage 1: A[m][0]=v[lane m], A[m][2]=v[lane 16+m], B=ones  => D[m][*]=v[m]+v[m+16]
// Fold the 8 accumulator VGPRs (rows) per lane, feed as B with A=ones:
// Stage 2: D2[m][n] = sum over all 16 rows => full 32-lane total in every element.
// Requires EXEC == all ones (call from uniform control flow, full waves).
__device__ __forceinline__ float wave_sum_wmma(float v) {
  v2f ones; ones[0] = 1.0f; ones[1] = 1.0f;
  v2f a;    a[0] = v;       a[1] = 0.0f;
  v8f c = {};
  c = __builtin_amdgcn_wmma_f32_16x16x4_f32(false, a, false, ones,
                                            (short)0, c, false, false);
  v2f b;
  b[0] = (c[0] + c[1]) + (c[2] + c[3]);
  b[1] = (c[4] + c[5]) + (c[6] + c[7]);
  v8f d = {};
  d = __builtin_amdgcn_wmma_f32_16x16x4_f32(false, ones, false, b,
                                            (short)0, d, false, false);
  return d[0];   // wave total, replicated to all lanes
}

// Deterministic block sum (result valid in thread 0 only).
__device__ __forceinline__ float block_sum(float v, float* red, int tid) {
  __syncthreads();                       // protect shared buffer reuse
  float w = wave_sum_wmma(v);            // wave32: 8 waves per 256-thread block
  if ((tid & 31) == 0) red[tid >> 5] = w;
  __syncthreads();
  float total = 0.0f;
  if (tid == 0) {
    #pragma unroll
    for (int i = 0; i < 8; ++i) total += red[i];
  }
  return total;
}

// ---- Kernel A: cross-entropy + smooth-L1 box loss (one ROI per thread) ----
__global__ void __launch_bounds__(256) cls_box_kernel(
    const float* __restrict__ class_logits,       // [4096, 81]
    const float* __restrict__ box_regression,     // [4096, 324]
    const float* __restrict__ regression_targets, // [4096, 4]
    const int*   __restrict__ labels,             // [4096]
    float* __restrict__ ws_cls,                   // [16] partials
    float* __restrict__ ws_box)                   // [16] partials
{
  __shared__ float red[8];
  const int tid = threadIdx.x;
  const int n   = blockIdx.x * 256 + tid;
  const int label = labels[n];

  const float* __restrict__ row = class_logits + (size_t)n * 81;
  float m = -INFINITY;
  for (int c = 0; c < 81; ++c) m = fmaxf(m, row[c]);
  float s = 0.0f;
  for (int c = 0; c < 81; ++c) s += expf(row[c] - m);
  const float logp  = (row[label] - m) - logf(s);
  const float cls_p = -logp;

  float box_p = 0.0f;
  const float* __restrict__ br = box_regression + (size_t)n * 324 + (size_t)label * 4;
  if (label > 0) {
    #pragma unroll
    for (int k = 0; k < 4; ++k) {
      float dlt = br[k] - regression_targets[(size_t)n * 4 + k];
      float ad  = fabsf(dlt);
      box_p += (ad < 1.0f) ? 0.5f * ad * ad : ad - 0.5f;
    }
  }

  const float ct = block_sum(cls_p, red, tid);   // uniform flow: EXEC all 1s
  const float bt = block_sum(box_p, red, tid);
  if (tid == 0) { ws_cls[blockIdx.x] = ct; ws_box[blockIdx.x] = bt; }
}

// ---- Kernel B: label-gathered mask channel -> sigmoid out + BCE reduce ----
__global__ void __launch_bounds__(256) mask_kernel(
    const float* __restrict__ mask_logits,   // [4096, 81, 784]
    const float* __restrict__ mask_targets,  // [4096, 784]
    const int*   __restrict__ labels,        // [4096]
    float* __restrict__ mask_prob,           // [4096, 784] (d_out + 3)
    float* __restrict__ ws_mask)             // [4096] partials
{
  __shared__ float red[8];
  const int tid = threadIdx.x;
  const int n   = blockIdx.x;
  const int label = labels[n];

  const float* __restrict__ ml = mask_logits + ((size_t)n * 81 + (size_t)label) * 784;
  const float* __restrict__ mt = mask_targets + (size_t)n * 784;
  float*       __restrict__ mp = mask_prob    + (size_t)n * 784;

  float p = 0.0f;
  for (int i = tid; i < 784; i += 256) {
    const float x = __builtin_nontemporal_load(ml + i);   // read-once gather
    const float t = __builtin_nontemporal_load(mt + i);
    const float sig = 1.0f / (1.0f + expf(-x));
    __builtin_nontemporal_store(sig, mp + i);             // write-once output
    // stable BCE-with-logits: max(x,0) - x*t + log1p(exp(-|x|))
    p += fmaxf(x, 0.0f) - x * t + log1pf(expf(-fabsf(x)));
  }

  const float tot = block_sum(p, red, tid);
  if (tid == 0) ws_mask[n] = tot;
}

// ---- Kernel C: final deterministic reductions -> 3 losses ----
__global__ void __launch_bounds__(256) finalize_kernel(
    const float* __restrict__ ws,   // [0..4095] mask, [4096..4111] cls, [4112..4127] box
    float* __restrict__ out)        // out[0..2] = losses
{
  __shared__ float red[8];
  const int tid = threadIdx.x;
  float p = 0.0f;
  for (int i = tid; i < 4096; i += 256) p += ws[i];
  const float mask_total = block_sum(p, red, tid);
  if (tid == 0) {
    float cls = 0.0f, box = 0.0f;
    #pragma unroll
    for (int k = 0; k < 16; ++k) { cls += ws[4096 + k]; box += ws[4112 + k]; }
    out[0] = cls * (1.0f / 4096.0f);
    out[1] = box * (1.0f / 4096.0f);
    out[2] = mask_total * (1.0f / (4096.0f * 784.0f));
  }
}

extern "C" void kernel_launch(void* const* d_in, const int* in_sizes, int n_in,
                              void* d_out, int out_size, void* d_ws, size_t ws_size,
                              hipStream_t stream) {
  const float* class_logits       = (const float*)d_in[0];
  const float* box_regression     = (const float*)d_in[1];
  const float* regression_targets = (const float*)d_in[2];
  const float* mask_logits        = (const float*)d_in[3];
  const float* mask_targets       = (const float*)d_in[4];
  const int*   labels             = (const int*)d_in[5];
  float* out = (float*)d_out;
  float* ws  = (float*)d_ws;     // needs 4128 floats; fully rewritten every launch

  const int N = 4096;
  cls_box_kernel<<<N / 256, 256, 0, stream>>>(class_logits, box_regression,
                                              regression_targets, labels,
                                              ws + 4096, ws + 4112);
  mask_kernel<<<N, 256, 0, stream>>>(mask_logits, mask_targets, labels,
                                     out + 3, ws);
  finalize_kernel<<<1, 256, 0, stream>>>(ws, out);
}